// BinarizableTree_59236188946450
// MI455X (gfx1250) — compile-verified
//
#include <hip/hip_runtime.h>
#include <hip/hip_bf16.h>
#include <math.h>

// ---------------------------------------------------------------------------
// BinarizableTree inside-algorithm for MI455X (gfx1250, wave32, WMMA).
//
// Shapes: N_TOK=40 (41 positions), BS=8, D=256, H=256, out dims O=2.
// Factorization:
//   Le = enc @ W1[:256], Re = enc @ W1[256:]          (WMMA fp32 GEMM)
//   M  = Le - Re                                       (elementwise)
//   S[i,j,k,b,:] = relu(M[j] + (Re[k]-Le[i]+b1)) @ W2 + b2   (WMMA fp32)
//   inside DP over widths 2..40 in LDS, single workgroup.
// ---------------------------------------------------------------------------

#define NPOS 41           // token positions 0..40
#define BS   8
#define D    256
#define H    256
#define NROW (NPOS * BS)  // 328 flattened (t,b) rows
#define NEGV (-1e30f)

typedef __attribute__((ext_vector_type(2))) float v2f;
typedef __attribute__((ext_vector_type(8))) float v8f;

__device__ __forceinline__ v8f wmma_f32_16x16x4(v2f a, v2f b, v8f c) {
  // D = A(16x4,f32) * B(4x16,f32) + C(16x16,f32); full fp32, matches reference.
  return __builtin_amdgcn_wmma_f32_16x16x4_f32(
      /*neg_a=*/false, a, /*neg_b=*/false, b,
      /*c_mod=*/(short)0, c, /*reuse_a=*/false, /*reuse_b=*/false);
}

// ---------------------------------------------------------------------------
// Kernel 1: LR[r, n] for r = t*8+b (328 rows, padded to 336), n in [0,512):
//   n <  256: Le[r, n]   = sum_d enc[r,d] * W1[d, n]
//   n >= 256: Re[r, n-256] = sum_d enc[r,d] * W1[256+d, n-256]
// One wave per 16x16 output tile; K-loop in steps of 4 (f32 WMMA).
// ---------------------------------------------------------------------------
__global__ void proj_kernel(const float* __restrict__ enc,
                            const float* __restrict__ W1,
                            float* __restrict__ LR) {
  const int m0   = blockIdx.x * 16;
  const int n0   = blockIdx.y * 16;
  const int lane = threadIdx.x;
  const int half = lane >> 4;     // 0: K=0..1, 1: K=2..3
  const int l15  = lane & 15;

  const int col = n0 + l15;                              // 0..511
  // B element (k, col): row stride 256 floats in the selected W1 half.
  const float* Wcol = (col < D) ? (W1 + col) : (W1 + D * H + (col - D));

  const int rowA   = m0 + l15;
  const int rA     = (rowA < NROW) ? rowA : (NROW - 1);  // clamp (no OOB read)
  const bool rvalid = (rowA < NROW);
  const float* encRow = enc + rA * D;

  v8f acc = {};
  for (int k0 = 0; k0 < D; k0 += 4) {
    const int kk = k0 + 2 * half;
    v2f a, b;
    float a0 = encRow[kk];
    float a1 = encRow[kk + 1];
    a[0] = rvalid ? a0 : 0.0f;
    a[1] = rvalid ? a1 : 0.0f;
    b[0] = Wcol[kk * H];
    b[1] = Wcol[(kk + 1) * H];
    acc = wmma_f32_16x16x4(a, b, acc);
  }
  // D layout: VGPR v -> rows (m0+v) lanes 0-15, (m0+v+8) lanes 16-31; N = l15.
  for (int v = 0; v < 8; ++v) {
    const int row = m0 + v + 8 * half;
    if (row < NROW) LR[row * 512 + col] = acc[v];
  }
}

// ---------------------------------------------------------------------------
// Kernel 2: M[r,h] = Le[r,h] - Re[r,h]
// ---------------------------------------------------------------------------
__global__ void mdiff_kernel(const float* __restrict__ LR,
                             float* __restrict__ M) {
  const int idx = blockIdx.x * blockDim.x + threadIdx.x;
  if (idx < NROW * H) {
    const int r = idx >> 8;
    const int h = idx & 255;
    M[idx] = LR[r * 512 + h] - LR[r * 512 + D + h];
  }
}

// ---------------------------------------------------------------------------
// Kernel 3: split scores. One wave per (i, k, b) with k >= i+2.
//   S[i,j,k,b,o] = sum_h relu(M[j,h] + cb[h]) * W2[h,o] + b2[o]
//   cb[h] = Re[k,h] - Le[i,h] + b1[h]   (staged in LDS, uniform over j)
// A-tile rows = split points j (up to 39, <=3 tiles of 16); B = W2 padded N->16.
// ---------------------------------------------------------------------------
__global__ void score_kernel(const float* __restrict__ LR,
                             const float* __restrict__ M,
                             const float* __restrict__ b1,
                             const float* __restrict__ W2,
                             const float* __restrict__ b2,
                             float* __restrict__ S) {
  const int i = blockIdx.x;
  const int k = blockIdx.y;
  const int b = blockIdx.z;
  if (k < i + 2) return;                 // uniform per block
  const int nj = k - i - 1;              // number of split points

  const int lane = threadIdx.x;
  const int half = lane >> 4;
  const int l15  = lane & 15;

  __shared__ float cb[H];
  __shared__ float w2s[H * 2];

  const int ri = i * BS + b;
  const int rk = k * BS + b;
  for (int h = lane; h < H; h += 32)
    cb[h] = LR[rk * 512 + D + h] - LR[ri * 512 + h] + b1[h];
  for (int h = lane; h < H * 2; h += 32)
    w2s[h] = W2[h];
  __syncthreads();

  const float b2n = (l15 < 2) ? b2[l15] : 0.0f;
  const int ntiles = (nj + 15) >> 4;

  for (int t = 0; t < ntiles; ++t) {
    const int m  = t * 16 + l15;         // A row (split index within span)
    const int j  = i + 1 + m;
    const int jc = (j <= NPOS - 1) ? j : (NPOS - 1);   // clamp (no OOB read)
    const bool valid = (m < nj);
    const float* Mj = M + (jc * BS + b) * H;

    v8f acc = {};
    for (int k0 = 0; k0 < H; k0 += 4) {
      const int kk = k0 + 2 * half;
      v2f a, bb;
      const float h0 = fmaxf(0.0f, Mj[kk]     + cb[kk]);
      const float h1 = fmaxf(0.0f, Mj[kk + 1] + cb[kk + 1]);
      a[0]  = valid ? h0 : 0.0f;         // select, keeps EXEC all-1s for WMMA
      a[1]  = valid ? h1 : 0.0f;
      bb[0] = (l15 < 2) ? w2s[kk * 2 + l15]       : 0.0f;
      bb[1] = (l15 < 2) ? w2s[(kk + 1) * 2 + l15] : 0.0f;
      acc = wmma_f32_16x16x4(a, bb, acc);
    }
    if (l15 < 2) {                       // only N=0,1 columns are real outputs
      for (int v = 0; v < 8; ++v) {
        const int mm = t * 16 + v + 8 * half;
        if (mm < nj) {
          const int jj = i + 1 + mm;
          S[(((i * NPOS + jj) * NPOS + k) * BS + b) * 2 + l15] = acc[v] + b2n;
        }
      }
    }
  }
}

// ---------------------------------------------------------------------------
// Kernel 4: full inside DP in one workgroup. Chart B (41*41*8 f32 = 52.5 KB)
// lives in LDS (CDNA5: up to 320 KB/WGP). 39 width steps, barrier-separated.
// Thread tid -> (i = tid/8, b = tid%8); online logsumexp over (j, o).
// Finally gathers out[b] = B[0, lengths[b], b].
// ---------------------------------------------------------------------------
__global__ void dp_kernel(const float* __restrict__ S,
                          const int* __restrict__ lengths,
                          float* __restrict__ out) {
  __shared__ float B[NPOS * NPOS * BS];
  const int tid = threadIdx.x;           // 320 threads = 10 waves

  for (int idx = tid; idx < NPOS * NPOS * BS; idx += 320) B[idx] = NEGV;
  for (int idx = tid; idx < (NPOS - 1) * BS; idx += 320) {
    const int t = idx >> 3, b = idx & 7;
    B[(t * NPOS + t + 1) * BS + b] = 0.0f;
  }
  __syncthreads();

  const int i = tid >> 3;
  const int b = tid & 7;
  const float2* S2 = reinterpret_cast<const float2*>(S);

  for (int w = 2; w <= NPOS - 1; ++w) {
    const bool active = (i + w <= NPOS - 1);
    float nb = NEGV;
    if (active) {
      const int k = i + w;
      float mx = -INFINITY, s = 0.0f;
      for (int j = i + 1; j < k; ++j) {
        const float base = B[(i * NPOS + j) * BS + b] + B[(j * NPOS + k) * BS + b];
        const float2 sc = S2[((i * NPOS + j) * NPOS + k) * BS + b];
        const float v0 = sc.x + base;
        const float v1 = sc.y + base;
        const float m2 = fmaxf(mx, fmaxf(v0, v1));
        s  = s * expf(mx - m2) + expf(v0 - m2) + expf(v1 - m2);
        mx = m2;
      }
      nb = mx + logf(s);
    }
    __syncthreads();                     // step w reads only widths < w
    if (active) B[(i * NPOS + (i + w)) * BS + b] = nb;
    __syncthreads();
  }

  if (tid < BS) {
    const int L = lengths[tid];
    out[tid] = B[(0 * NPOS + L) * BS + tid];
  }
}

// ---------------------------------------------------------------------------
// Launch. Workspace layout (bytes):
//   [0)                 LR : 328*512 f32  (671,744 B)
//   [671744)            M  : 328*256 f32  (335,872 B)
//   [1007616)           S  : 41*41*41*8*2 f32 (4,410,944 B)   total ~5.2 MB
// ---------------------------------------------------------------------------
extern "C" void kernel_launch(void* const* d_in, const int* in_sizes, int n_in,
                              void* d_out, int out_size, void* d_ws, size_t ws_size,
                              hipStream_t stream) {
  const float* enc     = (const float*)d_in[0];   // (41, 8, 256)
  const float* W1      = (const float*)d_in[1];   // (512, 256)
  const float* b1      = (const float*)d_in[2];   // (256,)
  const float* W2      = (const float*)d_in[3];   // (256, 2)
  const float* b2      = (const float*)d_in[4];   // (2,)
  const int*   lengths = (const int*)d_in[5];     // (8,)
  float* out = (float*)d_out;                     // (8,)

  char* ws = (char*)d_ws;
  float* LR = (float*)(ws);
  float* M  = (float*)(ws + (size_t)NROW * 512 * 4);
  float* S  = (float*)(ws + (size_t)NROW * 512 * 4 + (size_t)NROW * H * 4);

  // 1) Le/Re projection GEMM: 328(->336) x 512, K=256, fp32 WMMA.
  proj_kernel<<<dim3(21, 32), 32, 0, stream>>>(enc, W1, LR);
  // 2) M = Le - Re.
  mdiff_kernel<<<(NROW * H + 255) / 256, 256, 0, stream>>>(LR, M);
  // 3) Split scores S[i,j,k,b,0:2] for all i < j < k, fp32 WMMA.
  score_kernel<<<dim3(NPOS, NPOS, BS), 32, 0, stream>>>(LR, M, b1, W2, b2, S);
  // 4) Inside DP in LDS + gather.
  dp_kernel<<<1, 320, 0, stream>>>(S, lengths, out);
}